// CorticalMap_7370163880471
// MI455X (gfx1250) — compile-verified
//
#include <hip/hip_runtime.h>
#include <cmath>

// ---------------- static problem configuration ----------------
#define SS   128
#define LL   (SS*SS)          // 16384 units
#define AK   25
#define KA   (AK*AK)          // 625
#define LK   49
#define KL   (LK*LK)          // 2401
#define XW   (SS + AK - 1)    // 152 input width/height
#define NITERS 10
#define TW   56               // lateral LDS tile width: 8 units + 48 halo
#define PI_2F 1.57079632679489662f

typedef float v2f __attribute__((ext_vector_type(2)));
typedef float v8f __attribute__((ext_vector_type(8)));

// ---------------- device helpers ----------------
__device__ __forceinline__ int refl128(int i) {
    i = (i < 0) ? -i : i;
    return (i > SS - 1) ? (2 * (SS - 1) - i) : i;
}

__device__ __forceinline__ float wave_sum32(float v) {
    #pragma unroll
    for (int off = 16; off > 0; off >>= 1) v += __shfl_xor(v, off, 32);
    return v;
}

// ---------------- K0: build envelope tables in workspace ----------------
__global__ void cm_init_env(float* __restrict__ lri, float* __restrict__ aenv,
                            float* __restrict__ sre,
                            float lri_inv_max, float sre_inv_sum) {
    int t = blockIdx.x * blockDim.x + threadIdx.x;
    if (t < KL) {                       // long-range inhibition envelope (49x49)
        int ky = t / LK, kx = t % LK;
        float dx = (float)kx - 24.f, dy = (float)ky - 24.f;
        float r  = sqrtf(dx * dx + dy * dy);
        float im = 0.f;
        if (r < 2.5f)  { float c = __cosf(fminf(r * 0.2f,        1.f) * PI_2F); im = c * c; }
        float le = 0.f;
        if (r < 24.5f) { float c = __cosf(fminf(r * (1.f/49.f),  1.f) * PI_2F); le = c * c * (1.f - im); }
        lri[t] = le * lri_inv_max;
    }
    if (t < KA) {                       // afferent envelope (25x25), max == 1 at center
        int ky = t / AK, kx = t % AK;
        float dx = (float)kx - 12.f, dy = (float)ky - 12.f;
        float r  = sqrtf(dx * dx + dy * dy);
        float e  = 0.f;
        if (r < 12.5f) { float c = __cosf(fminf(r * 0.04f, 1.f) * PI_2F); e = c * c; }
        aenv[t] = e;
    }
    if (t < 25) {                       // 5x5 excitation kernel, sums to 1
        int ky = t / 5, kx = t % 5;
        float dx = (float)kx - 2.f, dy = (float)ky - 2.f;
        float r  = sqrtf(dx * dx + dy * dy);
        float s  = 0.f;
        if (r < 2.5f) { float c = __cosf(fminf(r * 0.2f, 1.f) * PI_2F); s = c * c; }
        sre[t] = s * sre_inv_sum;
    }
}

// ---------------- K1: state init ----------------
__global__ void cm_init_state(const float* __restrict__ last_lat,
                              const float* __restrict__ lat_mean0,
                              float* __restrict__ lat, float* __restrict__ lm) {
    int i = blockIdx.x * blockDim.x + threadIdx.x;
    if (i < LL) { lat[i] = last_lat[i]; lm[i] = lat_mean0[i]; }
}

// ---------------- K2: lateral weight row sums (one wave per unit) ----------------
__global__ void cm_lat_rowsum(const float* __restrict__ lw, float* __restrict__ invls) {
    int lane = threadIdx.x & 31, wave = threadIdx.x >> 5;
    int l = blockIdx.x * (blockDim.x >> 5) + wave;
    const float* row = lw + (size_t)l * KL;
    float s = 0.f;
    for (int k = lane; k < KL; k += 32) s += row[k];
    s = wave_sum32(s);
    if (lane == 0) invls[l] = 1.0f / s;
}

// ---------------- K3: afferent projection + x_tiles output ----------------
// one wave per unit; 8 consecutive units (same row) per block; staged x window in LDS
__global__ void cm_afferent(const float* __restrict__ x, const float* __restrict__ rfs,
                            const float* __restrict__ adat, const float* __restrict__ aenv_g,
                            float* __restrict__ rawaff_out, float* __restrict__ xtiles_out,
                            float* __restrict__ aff_ws) {
    __shared__ float xs[AK * 32];   // 25 rows x 32 cols window
    __shared__ float env[KA];
    int lane = threadIdx.x & 31, wave = threadIdx.x >> 5;
    int u0  = blockIdx.x * 8;
    int uy  = u0 / SS;
    int ux0 = u0 % SS;              // 8 | 128 so a block never straddles rows
    for (int i = threadIdx.x; i < AK * 32; i += blockDim.x) {
        int ry = i >> 5, rx = i & 31;
        xs[i] = x[(uy + ry) * XW + (ux0 + rx)];
    }
    for (int i = threadIdx.x; i < KA; i += blockDim.x) env[i] = aenv_g[i];
    __syncthreads();

    int l = u0 + wave;
    const float* rrow = rfs + (size_t)l * KA;
    float* xt = xtiles_out + (size_t)l * KA;
    float s1 = 0.f, s2 = 0.f;
    int ky = 0, kx = lane;
    while (kx >= AK) { kx -= AK; ++ky; }
    for (int k = lane; k < KA; k += 32) {
        float v = xs[ky * 32 + wave + kx] * env[k];
        xt[k] = v;                                   // coalesced x_tiles store
        float w = rrow[k];                           // coalesced 128B weight loads
        s1 += w;
        s2 = fmaf(v, w, s2);
        kx += 32; while (kx >= AK) { kx -= AK; ++ky; }
    }
    s1 = wave_sum32(s1);
    s2 = wave_sum32(s2);
    if (lane == 0) {
        float ratio = s2 / s1;             // = raw_aff / (Ka*FM)
        rawaff_out[l] = 62.5f * ratio;     // Ka*FM = 62.5
        aff_ws[l] = ratio - adat[l];
    }
}

// ---------------- K4: 5x5 reflect-pad smoothing conv via WMMA f32 16x16x4 ----------------
// one wave computes a 16-pixel row segment; 7 accumulating WMMAs cover 25 taps (pad to 28)
__global__ void cm_conv5_wmma(const float* __restrict__ lat, const float* __restrict__ sre_g,
                              float* __restrict__ lats) {
    int lane = threadIdx.x & 31, wave = threadIdx.x >> 5;
    int wg = blockIdx.x * (blockDim.x >> 5) + wave;   // 0..1023
    int y  = wg >> 3;
    int x0 = (wg & 7) << 4;
    int m     = lane & 15;              // output pixel M within tile
    int kHalf = (lane >> 4) << 1;       // lanes 0-15 hold K 0,1; lanes 16-31 hold K 2,3
    v8f c = {0.f, 0.f, 0.f, 0.f, 0.f, 0.f, 0.f, 0.f};
    #pragma unroll
    for (int chunk = 0; chunk < 7; ++chunk) {
        int t0 = chunk * 4 + kHalf;
        int t1 = t0 + 1;
        v2f a, b;
        if (t0 < 25) {
            a.x = lat[refl128(y + t0 / 5 - 2) * SS + refl128(x0 + m + t0 % 5 - 2)];
            b.x = sre_g[t0];
        } else { a.x = 0.f; b.x = 0.f; }
        if (t1 < 25) {
            a.y = lat[refl128(y + t1 / 5 - 2) * SS + refl128(x0 + m + t1 % 5 - 2)];
            b.y = sre_g[t1];
        } else { a.y = 0.f; b.y = 0.f; }
        // D[i,j] = sum_t A[i,t]*w_t (B columns identical) ; accumulate over chunks
        c = __builtin_amdgcn_wmma_f32_16x16x4_f32(false, a, false, b, (short)0, c,
                                                  false, false);
    }
    // C/D layout: lanes 0-15 carry rows 0..7 in c[0..7]; lanes 16-31 carry rows 8..15
    if (lane == 0) {
        #pragma unroll
        for (int i = 0; i < 8; ++i) lats[y * SS + x0 + i] = c[i];
    }
    if (lane == 16) {
        #pragma unroll
        for (int i = 0; i < 8; ++i) lats[y * SS + x0 + 8 + i] = c[i];
    }
}

// ---------------- K5: lateral inhibition + activation update ----------------
// one wave per unit; 8 units per block share a 49x56 LDS tile of lat_s (zero-padded)
__global__ void cm_lateral(const float* __restrict__ lats, const float* __restrict__ lw,
                           const float* __restrict__ invls, const float* __restrict__ aff,
                           const float* __restrict__ env_g,
                           float* __restrict__ lat, float* __restrict__ lm,
                           float* __restrict__ out_lat, int write_out) {
    __shared__ float tile[LK * TW];   // 49*56 floats
    __shared__ float env[KL];         // 2401 floats
    int lane = threadIdx.x & 31, wave = threadIdx.x >> 5;
    int u0  = blockIdx.x * 8;
    int uy  = u0 / SS;
    int ux0 = u0 % SS;
    for (int i = threadIdx.x; i < LK * TW; i += blockDim.x) {
        int ry = i / TW, rx = i - ry * TW;
        int gy = uy + ry - 24, gx = ux0 + rx - 24;
        tile[i] = (gy >= 0 && gy < SS && gx >= 0 && gx < SS) ? lats[gy * SS + gx] : 0.f;
    }
    for (int i = threadIdx.x; i < KL; i += blockDim.x) env[i] = env_g[i];
    __syncthreads();

    int l = u0 + wave;
    const float* wrow = lw + (size_t)l * KL;
    float acc = 0.f;
    int ky = 0, kx = lane;
    for (int k = lane; k < KL; k += 32) {
        __builtin_prefetch(wrow + k + 512, 0, 0);    // keep the weight stream ahead
        acc = fmaf(tile[ky * TW + wave + kx] * env[k], wrow[k], acc);
        kx += 32; while (kx >= LK) { kx -= LK; ++ky; }
    }
    acc = wave_sum32(acc);
    if (lane == 0) {
        float neg = acc * invls[l];                  // lat_w = w / sum(w)
        float v = lats[l] + aff[l] - 2.5f * neg;
        v = fmaxf(v, 0.f) * 2.2f;
        v = tanhf(v * 1.5f) * (1.f / 1.5f);
        lat[l] = v;
        lm[l]  = 0.5f * lm[l] + 0.5f * v;
        if (write_out) out_lat[l] = v;
    }
}

// ---------------- K6: Hebbian correlation partials ----------------
__global__ void cm_corr(const float* __restrict__ lmean, const float* __restrict__ lw,
                        const float* __restrict__ invls, const float* __restrict__ env_g,
                        float* __restrict__ part) {
    __shared__ float tile[LK * TW];
    __shared__ float env[KL];
    int lane = threadIdx.x & 31, wave = threadIdx.x >> 5;
    int u0  = blockIdx.x * 8;
    int uy  = u0 / SS;
    int ux0 = u0 % SS;
    for (int i = threadIdx.x; i < LK * TW; i += blockDim.x) {
        int ry = i / TW, rx = i - ry * TW;
        int gy = uy + ry - 24, gx = ux0 + rx - 24;
        tile[i] = (gy >= 0 && gy < SS && gx >= 0 && gx < SS) ? lmean[gy * SS + gx] : 0.f;
    }
    for (int i = threadIdx.x; i < KL; i += blockDim.x) env[i] = env_g[i];
    __syncthreads();

    int l = u0 + wave;
    const float* wrow = lw + (size_t)l * KL;
    float acc = 0.f;
    int ky = 0, kx = lane;
    for (int k = lane; k < KL; k += 32) {
        acc = fmaf(tile[ky * TW + wave + kx] * env[k], wrow[k], acc);
        kx += 32; while (kx >= LK) { kx -= LK; ++ky; }
    }
    acc = wave_sum32(acc);
    if (lane == 0) {
        // lat_n = w * (Kl*FM) / sum(w);  weighted row also scaled by lat_mean[l]
        part[l] = lmean[l] * 240.1f * invls[l] * acc;
    }
}

// ---------------- K7: deterministic fixed-order final reduction ----------------
__global__ void cm_reduce(const float* __restrict__ part, float* __restrict__ out_scalar) {
    __shared__ float sm[256];
    float s = 0.f;
    for (int i = threadIdx.x; i < LL; i += 256) s += part[i];
    sm[threadIdx.x] = s;
    __syncthreads();
    for (int off = 128; off > 0; off >>= 1) {
        if (threadIdx.x < off) sm[threadIdx.x] += sm[threadIdx.x + off];
        __syncthreads();
    }
    if (threadIdx.x == 0) out_scalar[0] = sm[0];
}

// ---------------- host launch ----------------
extern "C" void kernel_launch(void* const* d_in, const int* in_sizes, int n_in,
                              void* d_out, int out_size, void* d_ws, size_t ws_size,
                              hipStream_t stream) {
    (void)in_sizes; (void)n_in; (void)out_size; (void)ws_size;
    const float* x         = (const float*)d_in[0];
    const float* rfs       = (const float*)d_in[1];
    const float* lw        = (const float*)d_in[2];
    const float* adathresh = (const float*)d_in[3];
    const float* last_lat  = (const float*)d_in[4];
    const float* lat_mean0 = (const float*)d_in[5];
    float* out = (float*)d_out;

    // workspace layout (floats)
    float* ws     = (float*)d_ws;
    float* aff    = ws;                 // 16384
    float* lat    = ws + 16384;         // 16384
    float* lats   = ws + 32768;         // 16384
    float* lm     = ws + 49152;         // 16384
    float* invls  = ws + 65536;         // 16384
    float* part   = ws + 81920;         // 16384
    float* lri    = ws + 98304;         // 2401
    float* aenv   = ws + 98304 + 2432;  // 625
    float* sre    = ws + 98304 + 2432 + 640; // 25

    // output layout: raw_aff[16384] | lat[16384] | scalar[1] | x_tiles[16384*625]
    float* out_rawaff = out;
    float* out_lat    = out + 16384;
    float* out_scalar = out + 32768;
    float* out_xtiles = out + 32769;

    // host-side deterministic normalizers (mirror device envelope formulas)
    float lmax = 0.f;
    for (int ky = 0; ky < LK; ++ky) for (int kx = 0; kx < LK; ++kx) {
        float dx = (float)kx - 24.f, dy = (float)ky - 24.f;
        float r  = sqrtf(dx * dx + dy * dy);
        float im = 0.f;
        if (r < 2.5f)  { float c = cosf(fminf(r * 0.2f,       1.f) * PI_2F); im = c * c; }
        float le = 0.f;
        if (r < 24.5f) { float c = cosf(fminf(r * (1.f/49.f), 1.f) * PI_2F); le = c * c * (1.f - im); }
        if (le > lmax) lmax = le;
    }
    float ssum = 0.f;
    for (int ky = 0; ky < 5; ++ky) for (int kx = 0; kx < 5; ++kx) {
        float dx = (float)kx - 2.f, dy = (float)ky - 2.f;
        float r  = sqrtf(dx * dx + dy * dy);
        if (r < 2.5f) { float c = cosf(fminf(r * 0.2f, 1.f) * PI_2F); ssum += c * c; }
    }
    float lri_inv_max = 1.f / lmax;
    float sre_inv_sum = 1.f / ssum;

    cm_init_env<<<(KL + 255) / 256, 256, 0, stream>>>(lri, aenv, sre, lri_inv_max, sre_inv_sum);
    cm_init_state<<<LL / 256, 256, 0, stream>>>(last_lat, lat_mean0, lat, lm);
    cm_lat_rowsum<<<LL / 8, 256, 0, stream>>>(lw, invls);
    cm_afferent<<<LL / 8, 256, 0, stream>>>(x, rfs, adathresh, aenv,
                                            out_rawaff, out_xtiles, aff);
    for (int it = 0; it < NITERS; ++it) {
        cm_conv5_wmma<<<LL / 16 / 8, 256, 0, stream>>>(lat, sre, lats);
        cm_lateral<<<LL / 8, 256, 0, stream>>>(lats, lw, invls, aff, lri,
                                               lat, lm, out_lat, it == NITERS - 1);
    }
    cm_corr<<<LL / 8, 256, 0, stream>>>(lm, lw, invls, lri, part);
    cm_reduce<<<1, 256, 0, stream>>>(part, out_scalar);
}